// Outer_Product_Mean_84748294684976
// MI455X (gfx1250) — compile-verified
//
#include <hip/hip_runtime.h>

// Problem constants (from reference): B=1, S=128, R=256, C_M=256, C=32, C_Z=128
#define S_DIM 128
#define R_DIM 256
#define CM    256
#define C_DIM 32
#define CZ    128

typedef __attribute__((ext_vector_type(2))) float v2f;
typedef __attribute__((ext_vector_type(8))) float v8f;
typedef __attribute__((ext_vector_type(4))) unsigned int v4u;
typedef __attribute__((ext_vector_type(8))) int v8i;
typedef __attribute__((ext_vector_type(4))) int v4i;

__device__ __forceinline__ v8f wmma_f32(v2f a, v2f b, v8f c) {
  // 8 args: (neg_a, A, neg_b, B, c_mod, C, reuse_a, reuse_b)
  return __builtin_amdgcn_wmma_f32_16x16x4_f32(false, a, false, b, (short)0, c,
                                               false, false);
}

// -------------------------------------------------------------------------
// Kernel 1: y[r, c] = sum_k x[r, k] * W1[c, k] + b1[c]
//   r in [0, S*R)=32768, c in [0,32), k in [0,256)
// One wave per 16-row M-tile; N=32 = two 16-wide WMMA tiles; K chunked by 4.
// -------------------------------------------------------------------------
__global__ void __launch_bounds__(256)
opm_linear1(const float* __restrict__ x, const float* __restrict__ W1,
            const float* __restrict__ b1, float* __restrict__ y) {
  const int lane = threadIdx.x & 31;
  const int wave = threadIdx.x >> 5;
  const int mtile = blockIdx.x * 8 + wave;   // 2048 M-tiles total
  const int row0 = mtile * 16;
  const int half = lane >> 4;                // K sub-offset selector
  const int l16  = lane & 15;

  v8f acc0 = {};
  v8f acc1 = {};

  // A rows: M = l16 ; per-lane K = k0 + 2*half + {0,1}
  const float* xrow = x + (size_t)(row0 + l16) * CM;
  // B cols: N = l16 (+16); B[k,c] = W1[c,k] -> contiguous along k
  const float* w1a = W1 + (size_t)l16 * CM;
  const float* w1b = W1 + (size_t)(l16 + 16) * CM;

#pragma unroll 4
  for (int k0 = 0; k0 < CM; k0 += 4) {
    const int k = k0 + half * 2;
    v2f a;  a.x  = xrow[k]; a.y  = xrow[k + 1];
    v2f b0; b0.x = w1a[k];  b0.y = w1a[k + 1];
    v2f b1v; b1v.x = w1b[k]; b1v.y = w1b[k + 1];
    acc0 = wmma_f32(a, b0, acc0);
    acc1 = wmma_f32(a, b1v, acc1);
  }

  const float bias0 = b1[l16];
  const float bias1 = b1[l16 + 16];
#pragma unroll
  for (int v = 0; v < 8; ++v) {
    const int r = row0 + v + 8 * half;       // D row = v + 8*(lane/16)
    y[(size_t)r * C_DIM + l16]      = acc0[v] + bias0;
    y[(size_t)r * C_DIM + l16 + 16] = acc1[v] + bias1;
  }
}

// -------------------------------------------------------------------------
// Fused kernel: per block (i, j-tile of 32):
//  Stage:   TDM tensor_load_to_lds of Yi = y[:, i, :] (128x32 f32 tile)
//  Phase A: op_t[c][j][d] = sum_s Yi[s][c] * y[s, j0+j, d]
//           (GEMM M=32(c), N=1024(j,d), K=128(s)), result in LDS (128 KB)
//  Phase B: res[i, j0+j, z] = sum_{cd} op_t[c][j][d]*W2[z, c*32+d]/S + b2[z]
//           (GEMM M=32(j), N=128(z), K=1024(cd))
// LDS: Yi 16 KB + op_t 128 KB = 144 KB dynamic (CDNA5 WGP has 320 KB).
// -------------------------------------------------------------------------
__global__ void __launch_bounds__(256)
opm_fused(const float* __restrict__ y, const float* __restrict__ W2,
          const float* __restrict__ b2, float* __restrict__ res) {
  extern __shared__ float smem[];
  float* Yi  = smem;                     // [128][32]
  float* opt = smem + S_DIM * C_DIM;     // [32][32][32] = [c][j][d]

  const int i    = blockIdx.y;
  const int j0   = blockIdx.x * 32;
  const int lane = threadIdx.x & 31;
  const int wave = threadIdx.x >> 5;
  const int half = lane >> 4;
  const int l16  = lane & 15;

  // ---- Stage Yi via Tensor Data Mover (wave 0 only; EXEC-independent) ----
  // D# per CDNA5 ISA ch.8: 2D tile, tile_dim0=32 elems (c, contiguous),
  // tile_dim1=128 rows (s), row stride tensor_dim0_stride = R*C = 8192 elems,
  // data_size=2 (4 bytes), dest = LDS offset of Yi.
  if (wave == 0) {
    const unsigned long long ga =
        (unsigned long long)(const void*)(y + (size_t)i * C_DIM);
    const unsigned int lds_off = (unsigned int)(unsigned long long)(void*)Yi;

    v4u g0;
    g0[0] = 1u;                                  // count=1, user mode, no gather
    g0[1] = lds_off;                             // lds_addr [63:32]
    g0[2] = (unsigned int)ga;                    // global_addr low
    g0[3] = ((unsigned int)(ga >> 32) & 0x01FFFFFFu) | (2u << 30); // addr hi | type=2

    v8i g1;
    g1[0] = (int)(2u << 16);       // workgroup_mask=0 | data_size=2 (4B)
    g1[1] = (int)(32u << 16);      // tensor_dim0=32  -> bits[79:48] (low half)
    g1[2] = (int)(128u << 16);     // tensor_dim0 hi=0 | tensor_dim1=128 (low half)
    g1[3] = (int)(32u << 16);      // tensor_dim1 hi=0 | tile_dim0=32
    g1[4] = 128;                   // tile_dim1=128 | tile_dim2=0
    g1[5] = (int)(R_DIM * C_DIM);  // tensor_dim0_stride = 8192 (low 32 of 48)
    g1[6] = 0;                     // stride hi | tensor_dim1_stride low
    g1[7] = 0;

    v4i gz = {0, 0, 0, 0};
#if defined(__clang_major__) && (__clang_major__ >= 23)
    v8i gz8 = {0, 0, 0, 0, 0, 0, 0, 0};
    __builtin_amdgcn_tensor_load_to_lds(g0, g1, gz, gz, gz8, 0);
#else
    __builtin_amdgcn_tensor_load_to_lds(g0, g1, gz, gz, 0);
#endif
    __builtin_amdgcn_s_wait_tensorcnt(0);  // TDM writes to LDS complete
  }
  __syncthreads();

  // ---------------- Phase A ----------------
  {
    v8f acc[2][8];
#pragma unroll
    for (int m = 0; m < 2; ++m)
#pragma unroll
      for (int n = 0; n < 8; ++n) acc[m][n] = (v8f){0,0,0,0,0,0,0,0};

    const int ntbase = wave * 8;  // 8 waves x 8 N-tiles = 64 N-tiles (N=1024)

#pragma unroll 2
    for (int k0 = 0; k0 < S_DIM; k0 += 4) {
      const int s = k0 + half * 2;
      // A[c, s] = Yi[s][c]; two K values per lane
      v2f a0, a1;
      a0.x = Yi[s * C_DIM + l16];
      a0.y = Yi[(s + 1) * C_DIM + l16];
      a1.x = Yi[s * C_DIM + l16 + 16];
      a1.y = Yi[(s + 1) * C_DIM + l16 + 16];
#pragma unroll
      for (int n = 0; n < 8; ++n) {
        const int N  = (ntbase + n) * 16 + l16;   // 0..1023 = j*32 + d
        const int jj = j0 + (N >> 5);
        const int dd = N & 31;
        const float* bp = y + ((size_t)s * R_DIM + jj) * C_DIM + dd;
        v2f b;
        b.x = bp[0];
        b.y = bp[(size_t)R_DIM * C_DIM];          // s+1 plane
        acc[0][n] = wmma_f32(a0, b, acc[0][n]);
        acc[1][n] = wmma_f32(a1, b, acc[1][n]);
      }
    }

    // Scatter accumulators to LDS: op_t[c][N] with c = v + 8*half + 16*m
#pragma unroll
    for (int m = 0; m < 2; ++m)
#pragma unroll
      for (int n = 0; n < 8; ++n) {
        const int N = (ntbase + n) * 16 + l16;
#pragma unroll
        for (int v = 0; v < 8; ++v) {
          const int c = v + 8 * half + 16 * m;
          opt[c * 1024 + N] = acc[m][n][v];
        }
      }
  }
  __syncthreads();

  // ---------------- Phase B ----------------
  {
    // 16 (m,n)-tile combos, wave w owns combos 2w, 2w+1 (same m per wave)
    const int combo0 = wave * 2;
    const int mt  = combo0 >> 3;         // j-half
    const int nt0 = combo0 & 7;          // z-tile base
    const int jrow = l16 + 16 * mt;      // A row (local j)

    v8f acc0 = {};
    v8f acc1 = {};

#pragma unroll 4
    for (int k0 = 0; k0 < C_DIM * C_DIM; k0 += 4) {
      const int k = k0 + half * 2;       // k = c*32 + d; (k,k+1) same c
      const int c = k >> 5;
      const int d = k & 31;
      const float* ap = opt + c * 1024 + jrow * 32 + d;
      v2f a; a.x = ap[0]; a.y = ap[1];

      const int z0 = nt0 * 16 + l16;
      const float* bp0 = W2 + (size_t)z0 * (C_DIM * C_DIM) + k;
      const float* bp1 = W2 + (size_t)(z0 + 16) * (C_DIM * C_DIM) + k;
      v2f b0; b0.x = bp0[0]; b0.y = bp0[1];
      v2f b1v; b1v.x = bp1[0]; b1v.y = bp1[1];
      acc0 = wmma_f32(a, b0, acc0);
      acc1 = wmma_f32(a, b1v, acc1);
    }

    const float inv_s = 1.0f / (float)S_DIM;
    const int z0 = nt0 * 16 + l16;
    const float bz0 = b2[z0];
    const float bz1 = b2[z0 + 16];
#pragma unroll
    for (int v = 0; v < 8; ++v) {
      const int jl = v + 8 * half + 16 * mt;     // D row
      float* rp = res + ((size_t)i * R_DIM + (j0 + jl)) * CZ;
      rp[z0]      = acc0[v] * inv_s + bz0;
      rp[z0 + 16] = acc1[v] * inv_s + bz1;
    }
  }
}

// -------------------------------------------------------------------------
extern "C" void kernel_launch(void* const* d_in, const int* in_sizes, int n_in,
                              void* d_out, int out_size, void* d_ws,
                              size_t ws_size, hipStream_t stream) {
  const float* x  = (const float*)d_in[0];
  const float* W1 = (const float*)d_in[1];
  const float* b1 = (const float*)d_in[2];
  const float* W2 = (const float*)d_in[3];
  const float* b2 = (const float*)d_in[4];
  float* res = (float*)d_out;
  float* y   = (float*)d_ws;  // [S*R, 32] f32 = 4 MB scratch

  // y = x @ W1^T + b1
  opm_linear1<<<256, 256, 0, stream>>>(x, W1, b1, y);

  // Fused outer-product-mean + projection; 144 KB dynamic LDS per block
  const size_t lds_bytes = (size_t)(S_DIM * C_DIM + C_DIM * 32 * C_DIM) * 4;
  opm_fused<<<dim3(8, R_DIM), 256, lds_bytes, stream>>>(y, W2, b2, res);
}